// Net_84275848282728
// MI455X (gfx1250) — compile-verified
//
#include <hip/hip_runtime.h>
#include <math.h>

#define N_NODES 50000
#define F_IN    128
#define HC      256   // heads*hid = 4*64
#define NE      800000
#define BN_EPS  1e-5f

typedef float v2f __attribute__((ext_vector_type(2)));
typedef float v8f __attribute__((ext_vector_type(8)));

// ---------------- BatchNorm statistics ----------------
__global__ void bn_stats_kernel(const float* __restrict__ x, float* __restrict__ sums,
                                int n, int C, int rowsPer) {
    int c     = threadIdx.x % C;
    int rl    = threadIdx.x / C;
    int rstep = blockDim.x / C;
    int r0    = blockIdx.x * rowsPer;
    int rend  = min(n, r0 + rowsPer);
    float s = 0.f, s2 = 0.f;
    for (int r = r0 + rl; r < rend; r += rstep) {
        float v = x[(long)r * C + c];
        s += v; s2 += v * v;
    }
    atomicAdd(&sums[c], s);
    atomicAdd(&sums[C + c], s2);
}

__global__ void bn_finalize_kernel(const float* __restrict__ sums,
                                   const float* __restrict__ g, const float* __restrict__ b,
                                   float* __restrict__ scale, float* __restrict__ shift,
                                   int C, float invN) {
    int c = blockIdx.x * blockDim.x + threadIdx.x;
    if (c < C) {
        float mean = sums[c] * invN;
        float var  = sums[C + c] * invN - mean * mean;
        float sc   = g[c] * rsqrtf(var + BN_EPS);
        scale[c] = sc;
        shift[c] = b[c] - mean * sc;
    }
}

// ---------------- Fused BN-apply + dual GEMM via V_WMMA_F32_16X16X4_F32 ----------------
// Block: 256 threads (8 waves), 64-row band of A staged in LDS (33 KB of the 320 KB WGP pool).
// Wave w owns output columns [32w, 32w+32); each B fetch is reused across 4 row-bands:
// 16 WMMAs per 8 B-loads per k-step (4x better B reuse than a 16-row block).
__global__ __launch_bounds__(256) void gemm1_kernel(
    const float* __restrict__ x, const float* __restrict__ scale, const float* __restrict__ shift,
    const float* __restrict__ Wl, const float* __restrict__ bl, const float* __restrict__ Wg,
    float* __restrict__ lin, float* __restrict__ hg) {
    __shared__ float As[64][132];   // pad 128 -> 132 floats to spread LDS banks
    const int tid     = threadIdx.x;
    const int rowBase = blockIdx.x * 64;

    // Stage BN-normalized A tile: 64 rows x 128 cols, coalesced (2 rows per pass, tail-guarded)
    {
        int c  = tid & 127;
        int r0 = tid >> 7;      // 0..1
        float sc = scale[c], sh = shift[c];
        #pragma unroll
        for (int r = r0; r < 64; r += 2) {
            int row = rowBase + r;
            As[r][c] = (row < N_NODES) ? fmaf(x[(long)row * F_IN + c], sc, sh) : 0.f;
        }
    }
    __syncthreads();

    const int wave = tid >> 5, lane = tid & 31;
    const int m    = lane & 15;          // row (A) / col (B,C,D) within tile
    const int half = lane >> 4;          // 0: K=k,k+1   1: K=k+2,k+3
    const int col0 = wave * 32;
    const int n0   = col0 + m;
    const int n1   = col0 + 16 + m;

    v8f accL[4][2], accG[4][2];
    #pragma unroll
    for (int b = 0; b < 4; ++b) {
        accL[b][0] = {}; accL[b][1] = {};
        accG[b][0] = {}; accG[b][1] = {};
    }

    for (int k = 0; k < F_IN; k += 4) {
        const int kk = k + half * 2;
        // B: 4x16 f32 layout — lane holds W[kk][n], W[kk+1][n]; fetched once, used 4x
        v2f bL0, bL1, bG0, bG1;
        bL0.x = Wl[(long)kk * HC + n0]; bL0.y = Wl[(long)(kk + 1) * HC + n0];
        bL1.x = Wl[(long)kk * HC + n1]; bL1.y = Wl[(long)(kk + 1) * HC + n1];
        bG0.x = Wg[(long)kk * HC + n0]; bG0.y = Wg[(long)(kk + 1) * HC + n0];
        bG1.x = Wg[(long)kk * HC + n1]; bG1.y = Wg[(long)(kk + 1) * HC + n1];
        #pragma unroll
        for (int b = 0; b < 4; ++b) {
            // A: 16x4 f32 layout — lane m holds A[m][kk], A[m][kk+1]
            v2f a; a.x = As[b * 16 + m][kk]; a.y = As[b * 16 + m][kk + 1];
            accL[b][0] = __builtin_amdgcn_wmma_f32_16x16x4_f32(false, a, false, bL0, (short)0, accL[b][0], false, false);
            accL[b][1] = __builtin_amdgcn_wmma_f32_16x16x4_f32(false, a, false, bL1, (short)0, accL[b][1], false, false);
            accG[b][0] = __builtin_amdgcn_wmma_f32_16x16x4_f32(false, a, false, bG0, (short)0, accG[b][0], false, false);
            accG[b][1] = __builtin_amdgcn_wmma_f32_16x16x4_f32(false, a, false, bG1, (short)0, accG[b][1], false, false);
        }
    }

    // C/D layout: VGPR r -> row (r + 8*half), col m
    const float bb0 = bl[n0], bb1 = bl[n1];
    #pragma unroll
    for (int b = 0; b < 4; ++b) {
        #pragma unroll
        for (int r = 0; r < 8; ++r) {
            const long row = rowBase + b * 16 + r + half * 8;
            if (row < N_NODES) {
                lin[row * HC + n0] = accL[b][0][r] + bb0;
                lin[row * HC + n1] = accL[b][1][r] + bb1;
                hg [row * HC + n0] = accG[b][0][r];
                hg [row * HC + n1] = accG[b][1][r];
            }
        }
    }
}

// ---------------- Attention logits: wave-per-node, 8-lane head reductions ----------------
__global__ __launch_bounds__(256) void al_kernel(
    const float* __restrict__ hg, const float* __restrict__ asrc, const float* __restrict__ adst,
    float* __restrict__ als, float* __restrict__ ald) {
    const int wave = threadIdx.x >> 5, lane = threadIdx.x & 31;
    const int node = blockIdx.x * 8 + wave;
    if (node >= N_NODES) return;
    const float4* p   = (const float4*)(hg + (long)node * HC + lane * 8);
    const float4* as4 = (const float4*)(asrc + lane * 8);
    const float4* ad4 = (const float4*)(adst + lane * 8);
    float4 v0 = p[0],  v1 = p[1];
    float4 a0 = as4[0], a1 = as4[1];
    float4 d0 = ad4[0], d1 = ad4[1];
    float ps = v0.x*a0.x + v0.y*a0.y + v0.z*a0.z + v0.w*a0.w
             + v1.x*a1.x + v1.y*a1.y + v1.z*a1.z + v1.w*a1.w;
    float pd = v0.x*d0.x + v0.y*d0.y + v0.z*d0.z + v0.w*d0.w
             + v1.x*d1.x + v1.y*d1.y + v1.z*d1.z + v1.w*d1.w;
    // 8 lanes per head: reduce within the 8-lane partition (wave32)
    ps += __shfl_xor(ps, 1, 8); ps += __shfl_xor(ps, 2, 8); ps += __shfl_xor(ps, 4, 8);
    pd += __shfl_xor(pd, 1, 8); pd += __shfl_xor(pd, 2, 8); pd += __shfl_xor(pd, 4, 8);
    if ((lane & 7) == 0) {
        int h = lane >> 3;
        als[node * 4 + h] = ps;
        ald[node * 4 + h] = pd;
    }
}

// ---------------- Edge softmax numerator + denominator (layer 1) ----------------
__global__ void edge_sum_kernel(const int* __restrict__ src, const int* __restrict__ dst,
                                const float* __restrict__ als, const float* __restrict__ ald,
                                float* __restrict__ wbuf, float* __restrict__ ssum) {
    long i = (long)blockIdx.x * blockDim.x + threadIdx.x;
    if (i >= NE) return;
    const int s = src[i], d = dst[i];
    float4 es = *(const float4*)(als + (long)s * 4);
    float4 ed = *(const float4*)(ald + (long)d * 4);
    float e[4] = { es.x + ed.x, es.y + ed.y, es.z + ed.z, es.w + ed.w };
    #pragma unroll
    for (int h = 0; h < 4; ++h) {
        float t = e[h] > 0.f ? e[h] : 0.2f * e[h];   // leaky_relu(0.2)
        float w = expf(t);                           // logits bounded; max-shift unnecessary
        wbuf[i * 4 + h] = w;
        atomicAdd(&ssum[(long)d * 4 + h], w);
    }
}

// ---------------- Message scatter (layer 1): 64 threads/edge, float4 per thread ----------------
__global__ __launch_bounds__(256) void scatter1_kernel(
    const int* __restrict__ src, const int* __restrict__ dst,
    const float* __restrict__ hg, const float* __restrict__ wbuf, const float* __restrict__ ssum,
    float* __restrict__ out) {
    const long e  = (long)blockIdx.x * 4 + (threadIdx.x >> 6);
    const int  c4 = (threadIdx.x & 63) * 4;
    const int  s  = src[e], d = dst[e];
    const int  h  = c4 >> 6;
    const float alpha = wbuf[e * 4 + h] / (ssum[(long)d * 4 + h] + 1e-16f);
    float4 v = *(const float4*)(hg + (long)s * HC + c4);
    atomicAdd(&out[(long)d * HC + c4 + 0], v.x * alpha);
    atomicAdd(&out[(long)d * HC + c4 + 1], v.y * alpha);
    atomicAdd(&out[(long)d * HC + c4 + 2], v.z * alpha);
    atomicAdd(&out[(long)d * HC + c4 + 3], v.w * alpha);
}

// ---------------- h2 = relu(lin + gat1 + bias), in place over lin ----------------
__global__ void h2_kernel(float* __restrict__ lin, const float* __restrict__ gat,
                          const float* __restrict__ bias) {
    long i = (long)blockIdx.x * blockDim.x + threadIdx.x;
    if (i >= (long)N_NODES * HC) return;
    int c = (int)(i & (HC - 1));
    float v = lin[i] + gat[i] + bias[c];
    lin[i] = v > 0.f ? v : 0.f;
}

// ---------------- Layer 2: BN-apply + dual GEMV (256 -> 2), wave per node ----------------
__global__ __launch_bounds__(256) void layer2_kernel(
    const float* __restrict__ h2, const float* __restrict__ scale, const float* __restrict__ shift,
    const float* __restrict__ W3, const float* __restrict__ b3,
    const float* __restrict__ Wg2, const float* __restrict__ asrc2, const float* __restrict__ adst2,
    float* __restrict__ lin3, float* __restrict__ hg2,
    float* __restrict__ al2s, float* __restrict__ al2d) {
    const int wave = threadIdx.x >> 5, lane = threadIdx.x & 31;
    const int node = blockIdx.x * 8 + wave;
    if (node >= N_NODES) return;
    const float4* p = (const float4*)(h2 + (long)node * HC + lane * 8);
    float4 v0 = p[0], v1 = p[1];
    float vv[8] = { v0.x, v0.y, v0.z, v0.w, v1.x, v1.y, v1.z, v1.w };
    float accL0 = 0.f, accL1 = 0.f, accG0 = 0.f, accG1 = 0.f;
    #pragma unroll
    for (int j = 0; j < 8; ++j) {
        int cc  = lane * 8 + j;
        float v = fmaf(vv[j], scale[cc], shift[cc]);
        accL0 = fmaf(v, W3 [cc * 2 + 0], accL0);
        accL1 = fmaf(v, W3 [cc * 2 + 1], accL1);
        accG0 = fmaf(v, Wg2[cc * 2 + 0], accG0);
        accG1 = fmaf(v, Wg2[cc * 2 + 1], accG1);
    }
    #pragma unroll
    for (int mk = 16; mk >= 1; mk >>= 1) {
        accL0 += __shfl_xor(accL0, mk, 32);
        accL1 += __shfl_xor(accL1, mk, 32);
        accG0 += __shfl_xor(accG0, mk, 32);
        accG1 += __shfl_xor(accG1, mk, 32);
    }
    if (lane == 0) {
        lin3[node * 2 + 0] = accL0 + b3[0];
        lin3[node * 2 + 1] = accL1 + b3[1];
        hg2 [node * 2 + 0] = accG0;
        hg2 [node * 2 + 1] = accG1;
        al2s[node] = accG0 * asrc2[0] + accG1 * asrc2[1];
        al2d[node] = accG0 * adst2[0] + accG1 * adst2[1];
    }
}

// ---------------- Edge softmax (layer 2, single head) ----------------
__global__ void edge2_sum_kernel(const int* __restrict__ src, const int* __restrict__ dst,
                                 const float* __restrict__ al2s, const float* __restrict__ al2d,
                                 float* __restrict__ wbuf2, float* __restrict__ s2sum) {
    long i = (long)blockIdx.x * blockDim.x + threadIdx.x;
    if (i >= NE) return;
    const int s = src[i], d = dst[i];
    float e = al2s[s] + al2d[d];
    e = e > 0.f ? e : 0.2f * e;
    float w = expf(e);
    wbuf2[i] = w;
    atomicAdd(&s2sum[d], w);
}

__global__ void scatter2_kernel(const int* __restrict__ src, const int* __restrict__ dst,
                                const float* __restrict__ hg2, const float* __restrict__ wbuf2,
                                const float* __restrict__ s2sum, float* __restrict__ gat2) {
    long i = (long)blockIdx.x * blockDim.x + threadIdx.x;
    if (i >= NE) return;
    const int s = src[i], d = dst[i];
    const float alpha = wbuf2[i] / (s2sum[d] + 1e-16f);
    atomicAdd(&gat2[(long)d * 2 + 0], hg2[(long)s * 2 + 0] * alpha);
    atomicAdd(&gat2[(long)d * 2 + 1], hg2[(long)s * 2 + 1] * alpha);
}

__global__ void final_kernel(const float* __restrict__ lin3, const float* __restrict__ gat2,
                             const float* __restrict__ b2, float* __restrict__ out) {
    long i = (long)blockIdx.x * blockDim.x + threadIdx.x;
    if (i >= (long)N_NODES * 2) return;
    float v = lin3[i] + gat2[i] + b2[i & 1];
    out[i] = v > 0.f ? v : 0.f;
}

// ---------------- Host orchestration ----------------
extern "C" void kernel_launch(void* const* d_in, const int* in_sizes, int n_in,
                              void* d_out, int out_size, void* d_ws, size_t ws_size,
                              hipStream_t stream) {
    const float* x       = (const float*)d_in[0];
    const int*   edge    = (const int*)d_in[1];      // [2, E] int32 (JAX canonicalized)
    const int*   src     = edge;
    const int*   dst     = edge + NE;
    const float* bn1_g   = (const float*)d_in[2];
    const float* bn1_b   = (const float*)d_in[3];
    const float* bn3_g   = (const float*)d_in[4];
    const float* bn3_b   = (const float*)d_in[5];
    const float* lin1_W  = (const float*)d_in[6];
    const float* lin1_b  = (const float*)d_in[7];
    const float* lin3_W  = (const float*)d_in[8];
    const float* lin3_b  = (const float*)d_in[9];
    const float* g1_W    = (const float*)d_in[10];
    const float* g1_asrc = (const float*)d_in[11];
    const float* g1_adst = (const float*)d_in[12];
    const float* g1_bias = (const float*)d_in[13];
    const float* g2_W    = (const float*)d_in[14];
    const float* g2_asrc = (const float*)d_in[15];
    const float* g2_adst = (const float*)d_in[16];
    const float* g2_bias = (const float*)d_in[17];

    // Workspace carve-out (floats)
    float* ws = (float*)d_ws;
    float* lin    = ws; ws += (size_t)N_NODES * HC;   // also reused as h2 (in-place relu)
    float* hg     = ws; ws += (size_t)N_NODES * HC;
    float* gat1   = ws; ws += (size_t)N_NODES * HC;
    float* als    = ws; ws += (size_t)N_NODES * 4;
    float* ald    = ws; ws += (size_t)N_NODES * 4;
    float* ssum   = ws; ws += (size_t)N_NODES * 4;
    float* wbuf   = ws; ws += (size_t)NE * 4;
    float* stats1 = ws; ws += 2 * F_IN;
    float* scale1 = ws; ws += F_IN;
    float* shift1 = ws; ws += F_IN;
    float* stats3 = ws; ws += 2 * HC;
    float* scale3 = ws; ws += HC;
    float* shift3 = ws; ws += HC;
    float* lin3   = ws; ws += (size_t)N_NODES * 2;
    float* hg2    = ws; ws += (size_t)N_NODES * 2;
    float* al2s   = ws; ws += (size_t)N_NODES;
    float* al2d   = ws; ws += (size_t)N_NODES;
    float* s2sum  = ws; ws += (size_t)N_NODES;
    float* wbuf2  = ws; ws += (size_t)NE;
    float* gat2   = ws; ws += (size_t)N_NODES * 2;

    // Zero accumulators (capturable stream ops)
    hipMemsetAsync(stats1, 0, 2 * F_IN * sizeof(float), stream);
    hipMemsetAsync(stats3, 0, 2 * HC * sizeof(float), stream);
    hipMemsetAsync(gat1,  0, (size_t)N_NODES * HC * sizeof(float), stream);
    hipMemsetAsync(ssum,  0, (size_t)N_NODES * 4 * sizeof(float), stream);
    hipMemsetAsync(s2sum, 0, (size_t)N_NODES * sizeof(float), stream);
    hipMemsetAsync(gat2,  0, (size_t)N_NODES * 2 * sizeof(float), stream);

    // Layer 1
    bn_stats_kernel<<<(N_NODES + 255) / 256, 256, 0, stream>>>(x, stats1, N_NODES, F_IN, 256);
    bn_finalize_kernel<<<1, F_IN, 0, stream>>>(stats1, bn1_g, bn1_b, scale1, shift1, F_IN, 1.f / N_NODES);
    gemm1_kernel<<<(N_NODES + 63) / 64, 256, 0, stream>>>(x, scale1, shift1, lin1_W, lin1_b, g1_W, lin, hg);
    al_kernel<<<(N_NODES + 7) / 8, 256, 0, stream>>>(hg, g1_asrc, g1_adst, als, ald);
    edge_sum_kernel<<<(NE + 255) / 256, 256, 0, stream>>>(src, dst, als, ald, wbuf, ssum);
    scatter1_kernel<<<NE / 4, 256, 0, stream>>>(src, dst, hg, wbuf, ssum, gat1);
    h2_kernel<<<((long)N_NODES * HC + 255) / 256, 256, 0, stream>>>(lin, gat1, g1_bias);

    // Layer 2
    bn_stats_kernel<<<(N_NODES + 255) / 256, 256, 0, stream>>>(lin, stats3, N_NODES, HC, 256);
    bn_finalize_kernel<<<1, HC, 0, stream>>>(stats3, bn3_g, bn3_b, scale3, shift3, HC, 1.f / N_NODES);
    layer2_kernel<<<(N_NODES + 7) / 8, 256, 0, stream>>>(lin, scale3, shift3, lin3_W, lin3_b,
                                                         g2_W, g2_asrc, g2_adst, lin3, hg2, al2s, al2d);
    edge2_sum_kernel<<<(NE + 255) / 256, 256, 0, stream>>>(src, dst, al2s, al2d, wbuf2, s2sum);
    scatter2_kernel<<<(NE + 255) / 256, 256, 0, stream>>>(src, dst, hg2, wbuf2, s2sum, gat2);
    final_kernel<<<((long)N_NODES * 2 + 255) / 256, 256, 0, stream>>>(lin3, gat2, g2_bias, (float*)d_out);
}